// Sparsecone_61856118997459
// MI455X (gfx1250) — compile-verified
//
#include <hip/hip_runtime.h>

// Sparsemax-with-cone-rescale, row-wise over [32768, 2048] fp32.
// One wave32 per row, 64 elements/lane in VGPRs. Rows are double-buffered
// through LDS with CDNA5 async global->LDS loads (ASYNCcnt). The sparsemax
// threshold is found exactly via support-set compaction (ballot prefix into a
// 32-float LDS slot; support is {z > max-1}, ~13/2048 elems for Gaussian data)
// followed by Newton iteration t <- (S(t)-1)/K(t), which converges monotonically
// for the convex piecewise-linear f(t) = sum(relu(z-t)).

#define ROWS 32768
#define DIM  2048
#define NWG  4096            // persistent single-wave workgroups; 8 rows each
#define EPT  64              // elements per lane (DIM / 32)
#define NCH  16              // float4 chunks per lane (EPT / 4)
#define ROW_BYTES (DIM * 4)  // 8192

typedef __attribute__((ext_vector_type(4))) float v4f;

__device__ __forceinline__ float wave_sum(float v) {
#pragma unroll
    for (int m = 16; m > 0; m >>= 1) v += __shfl_xor(v, m, 32);
    return v;
}

__device__ __forceinline__ float wave_max(float v) {
#pragma unroll
    for (int m = 16; m > 0; m >>= 1) v = fmaxf(v, __shfl_xor(v, m, 32));
    return v;
}

__global__ __launch_bounds__(32)
void sparsecone_kernel(const float* __restrict__ x, float* __restrict__ out) {
    __shared__ __align__(16) float sbuf[2][DIM];   // 16 KB double buffer
    __shared__ float s_cand[32];                   // compacted support candidates

    const unsigned lane = threadIdx.x;             // 0..31, wave32
    const int      wg   = blockIdx.x;

    // LDS byte addresses (low 32 bits of the flat LDS address = LDS offset).
    const unsigned ldsBase[2] = {
        (unsigned)(size_t)(&sbuf[0][0]) + lane * 16u,
        (unsigned)(size_t)(&sbuf[1][0]) + lane * 16u
    };

    // Issue a full-row async copy (16 x 512B = 8KB) global -> LDS buffer b.
    // GVS mode: mem_addr = SADDR(64b sgpr pair) + VADDR[31:0]; the same lane
    // VGPR pattern addresses both the global source and the LDS destination.
    // Input is streamed exactly once -> non-temporal hint.
    auto issue_row = [&](int row, int b) {
        const unsigned goff  = (unsigned)row * (unsigned)ROW_BYTES + lane * 16u;
        const unsigned laddr = ldsBase[b];
#pragma unroll
        for (int c = 0; c < NCH; ++c) {
            asm volatile("global_load_async_to_lds_b128 %0, %1, %2 th:TH_LOAD_NT"
                         :
                         : "v"(laddr + (unsigned)(c * 512)),
                           "v"(goff  + (unsigned)(c * 512)),
                           "s"(x)
                         : "memory");
        }
    };

    int row = wg;
    if (row < ROWS) issue_row(row, 0);

    int b = 0;
    for (; row < ROWS; row += NWG, b ^= 1) {
        const int nxt = row + NWG;
        if (nxt < ROWS) {
            issue_row(nxt, b ^ 1);                             // prefetch next row
            asm volatile("s_wait_asynccnt 0x10" ::: "memory"); // current 16 done
        } else {
            asm volatile("s_wait_asynccnt 0x0"  ::: "memory"); // drain
        }

        // ---- LDS -> registers (16x ds_load_b128) ----
        float z[EPT];
        const v4f* p = reinterpret_cast<const v4f*>(&sbuf[b][0]);
#pragma unroll
        for (int c = 0; c < NCH; ++c) {
            v4f q = p[c * 32 + lane];
            z[4 * c + 0] = q.x; z[4 * c + 1] = q.y;
            z[4 * c + 2] = q.z; z[4 * c + 3] = q.w;
        }

        // ---- cone rescale: alpha = (1 + dim*q) / (sum + dim*q) ----
        float s = 0.0f;
#pragma unroll
        for (int e = 0; e < EPT; ++e) s += z[e];
        s = wave_sum(s);
        const float dq = 204.8f;                 // dim * q = 2048 * 0.1
        const float alpha = (1.0f + dq) / (s + dq);
#pragma unroll
        for (int e = 0; e < EPT; ++e) z[e] *= alpha;

        // ---- row max; threshold tau lives in [m-1, m) ----
        float m = -3.402823466e+38f;
#pragma unroll
        for (int e = 0; e < EPT; ++e) m = fmaxf(m, z[e]);
        m = wave_max(m);
        const float thr = m - 1.0f;              // support subset of {z > thr}

        // ---- compact support candidates {z > thr} into LDS (ballot prefix) ----
        unsigned cnt = 0;                        // uniform (sgpr) count
#pragma unroll
        for (int e = 0; e < EPT; ++e) {
            const bool g = z[e] > thr;
            const unsigned long long ball = __ballot(g);  // wave32: low 32 bits
            if (ball) {                                    // uniform; usually false
                if (g) {
                    const unsigned off =
                        cnt + (unsigned)__popcll(ball & ((1ull << lane) - 1ull));
                    if (off < 32u) s_cand[off] = z[e];
                }
                cnt += (unsigned)__popcll(ball);
            }
        }

        // ---- Newton: t <- (S(t)-1)/K(t) over {z > t}; monotone, exact on
        //      convergence of the support set. t0 = thr has f(t0) >= 1. ----
        float tau;
        if (cnt <= 32u) {                        // fast path: 1 candidate/lane
            const float c = (lane < cnt) ? s_cand[lane] : -3.0e+38f;
            float t = thr;
            for (int it = 0; it < 8; ++it) {
                const bool g = c > t;
                const float k  = wave_sum(g ? 1.0f : 0.0f);
                const float s2 = wave_sum(g ? c  : 0.0f);
                t = (s2 - 1.0f) / k;             // k >= 1: max cand m > t always
            }
            tau = t;
        } else {                                 // ~impossible for Gaussian rows
            float t = thr;
            for (int it = 0; it < 16; ++it) {
                float k = 0.0f, s2 = 0.0f;
#pragma unroll
                for (int e = 0; e < EPT; ++e) {
                    if (z[e] > t) { k += 1.0f; s2 += z[e]; }
                }
                k  = wave_sum(k);
                s2 = wave_sum(s2);
                t = (s2 - 1.0f) / k;
            }
            tau = t;
        }

        // ---- prob = relu(z - tau); streaming non-temporal stores ----
        v4f* o = reinterpret_cast<v4f*>(out + (size_t)row * DIM);
#pragma unroll
        for (int c = 0; c < NCH; ++c) {
            v4f q;
            q.x = fmaxf(z[4 * c + 0] - tau, 0.0f);
            q.y = fmaxf(z[4 * c + 1] - tau, 0.0f);
            q.z = fmaxf(z[4 * c + 2] - tau, 0.0f);
            q.w = fmaxf(z[4 * c + 3] - tau, 0.0f);
            __builtin_nontemporal_store(q, o + c * 32 + lane);
        }
    }
}

extern "C" void kernel_launch(void* const* d_in, const int* in_sizes, int n_in,
                              void* d_out, int out_size, void* d_ws, size_t ws_size,
                              hipStream_t stream) {
    (void)in_sizes; (void)n_in; (void)d_ws; (void)ws_size; (void)out_size;
    const float* x  = (const float*)d_in[0];
    float*      out = (float*)d_out;
    sparsecone_kernel<<<NWG, 32, 0, stream>>>(x, out);
}